// GenerateOneStageDetections_32555852104108
// MI455X (gfx1250) — compile-verified
//
#include <hip/hip_runtime.h>
#include <hip/hip_bf16.h>
#include <cstdint>

#ifndef __has_builtin
#define __has_builtin(x) 0
#endif

#define BDIM      256
#define BATCH     2
#define NCLS      90          // foreground classes (drop class 0)
#define NTOT      76725       // sum of level anchor counts
#define CAP       2048        // per-(b,c) candidate cap (approx top-k)
#define CHUNK     1024        // column staging chunk (floats); x2 buffers
#define NBINS     128
#define MAXT      100
#define FLATN     (NCLS * MAXT)   // 9000
#define IOU_THR_F 0.3f
#define LOGIT_THR (-2.9444389791664403f)   // logit(0.05)
#define BBOX_CLIPF 4.135166556742356f      // log(1000/16)
#define NEG_SENT  (-1.0e30f)

constexpr int LVL_N[5]   = {57600, 14400, 3600, 900, 225};
constexpr int LVL_OFF[5] = {0, 57600, 72000, 75600, 76500};
constexpr int nchunks_of(int n) { return (n + CHUNK - 1) / CHUNK; }
constexpr int NCHUNKS_TOT = nchunks_of(57600) + nchunks_of(14400) +
                            nchunks_of(3600) + nchunks_of(900) + nchunks_of(225);

struct PtrPack {
  const float* box[5];
  const float* cls[5];
  const float* anc[5];
};

// ---------- CDNA5 feature detection (per-pass; host pass falls back) ----------
#if __has_builtin(__builtin_amdgcn_tensor_load_to_lds) && __has_builtin(__builtin_amdgcn_s_wait_tensorcnt)
#define HAVE_TDM 1
#else
#define HAVE_TDM 0
#endif

#if __has_builtin(__builtin_amdgcn_global_load_async_to_lds_b128) && __has_builtin(__builtin_amdgcn_s_wait_asynccnt)
#define HAVE_ASYNC 1
#else
#define HAVE_ASYNC 0
#endif

// gcc-style vector typedefs (matches clang's builtin signature spelling)
typedef unsigned int tdm_v4u __attribute__((vector_size(16)));
typedef int          tdm_v8i __attribute__((vector_size(32)));
typedef int          tdm_v4i __attribute__((vector_size(16)));

typedef int v4i_vs __attribute__((vector_size(16)));
typedef __attribute__((address_space(1))) v4i_vs as1_v4i;   // AS(1): '__device__'
typedef __attribute__((address_space(3))) v4i_vs as3_v4i;   // AS(3): '__shared__'

__device__ __forceinline__ unsigned ordf(float f) {
  unsigned u = __float_as_uint(f);
  return (u & 0x80000000u) ? ~u : (u | 0x80000000u);   // monotone float->uint
}

// Map flat chunk index -> (level, row offset within level, rows in chunk)
__device__ __forceinline__ void chunk_of(int ci, int& l, int& j0, int& rows) {
  l = 0; j0 = 0; rows = 0;
#pragma unroll
  for (int q = 0; q < 5; ++q) {
    int nc = (LVL_N[q] + CHUNK - 1) / CHUNK;
    if (ci < nc) {
      l = q;
      j0 = ci * CHUNK;
      rows = LVL_N[q] - j0;
      if (rows > CHUNK) rows = CHUNK;
      return;
    }
    ci -= nc;
  }
}

// Issue DMA of one strided class column chunk (stride 91 floats) into LDS.
// TDM: D# = 1 x rows tile, tensor_dim0_stride = 91 elements; gathers the
// column contiguously into LDS. Fallback: synchronous cooperative copy.
__device__ __forceinline__ void stage_issue(const float* colbase, int rows,
                                            float* stage, int tid) {
#if HAVE_TDM
  if (tid == 0) {
    unsigned long long ga = (unsigned long long)(uintptr_t)colbase;
    unsigned lds_off = (unsigned)(uintptr_t)stage;  // low 32 bits of flat LDS ptr = LDS byte offset
    tdm_v4u g0;
    g0[0] = 1u;                                        // count=1, no gather
    g0[1] = lds_off;                                   // lds_addr
    g0[2] = (unsigned)(ga & 0xFFFFFFFFull);            // global_addr[31:0]
    g0[3] = (unsigned)((ga >> 32) & 0x01FFFFFFull) | (2u << 30);  // addr[56:32] | type=2
    tdm_v8i g1;
    g1[0] = 0x00020000;                                // wg_mask=0, data_size=2 (4B)
    g1[1] = 0x00010000;                                // tensor_dim0 = 1 (lo16 at [31:16])
    g1[2] = (int)(((unsigned)rows & 0xFFFFu) << 16);   // tensor_dim1 lo16
    g1[3] = 0x00010000;                                // tile_dim0 = 1
    g1[4] = rows & 0xFFFF;                             // tile_dim1 = rows, tile_dim2 = 0
    g1[5] = 91;                                        // tensor_dim0_stride lo32 (elements)
    g1[6] = 0;
    g1[7] = 0;
    tdm_v4i g2 = {0, 0, 0, 0};
    tdm_v4i g3 = {0, 0, 0, 0};
    tdm_v8i g4 = {0, 0, 0, 0, 0, 0, 0, 0};
    __builtin_amdgcn_tensor_load_to_lds(g0, g1, g2, g3, g4, 0);
  }
#else
  for (int k = tid; k < rows; k += BDIM) stage[k] = colbase[(size_t)k * 91];
#endif
}

__device__ __forceinline__ void stage_wait_keep1() {   // all but newest DMA done
#if HAVE_TDM
  __builtin_amdgcn_s_wait_tensorcnt(1);
#endif
}
__device__ __forceinline__ void stage_wait_all() {
#if HAVE_TDM
  __builtin_amdgcn_s_wait_tensorcnt(0);
#endif
}

__device__ __forceinline__ unsigned long long
block_argmax(unsigned long long key, unsigned long long* s_red,
             unsigned long long* s_best, int tid) {
#pragma unroll
  for (int m = 16; m > 0; m >>= 1) {
    unsigned long long o = __shfl_xor(key, m, 32);
    if (o > key) key = o;
  }
  if ((tid & 31) == 0) s_red[tid >> 5] = key;
  __syncthreads();
  if (tid == 0) {
    unsigned long long best = s_red[0];
#pragma unroll
    for (int w = 1; w < BDIM / 32; ++w)
      if (s_red[w] > best) best = s_red[w];
    *s_best = best;
  }
  __syncthreads();
  unsigned long long r = *s_best;
  __syncthreads();
  return r;
}

// ---------------- Kernel 1: decode + clip boxes -> ws ----------------
__global__ __launch_bounds__(BDIM) void k_decode(PtrPack p, const float* img,
                                                 float* boxes_out) {
  int t = blockIdx.x * blockDim.x + threadIdx.x;
  if (t >= BATCH * NTOT) return;
  int b = t / NTOT;
  int i = t - b * NTOT;
  int l = 0;
#pragma unroll
  for (int q = 1; q < 5; ++q)
    if (i >= LVL_OFF[q]) l = q;
  int j = i - LVL_OFF[l];
  const float* e = p.box[l] + ((size_t)b * LVL_N[l] + j) * 4;
  const float* a = p.anc[l] + ((size_t)b * LVL_N[l] + j) * 4;
  float dy = e[0], dx = e[1];
  float dh = fminf(e[2], BBOX_CLIPF), dw = fminf(e[3], BBOX_CLIPF);
  float ah = a[2] - a[0] + 1.0f, aw = a[3] - a[1] + 1.0f;
  float ayc = a[0] + 0.5f * ah, axc = a[1] + 0.5f * aw;
  float yc = dy * ah + ayc, xc = dx * aw + axc;
  float h = __expf(dh) * ah, w = __expf(dw) * aw;
  float y1 = yc - 0.5f * h, x1 = xc - 0.5f * w;
  float y2 = y1 + h - 1.0f, x2 = x1 + w - 1.0f;
  float H = img[b * 2 + 0], W = img[b * 2 + 1];
  y1 = fminf(fmaxf(y1, 0.f), H);
  x1 = fminf(fmaxf(x1, 0.f), W);
  y2 = fminf(fmaxf(y2, 0.f), H);
  x2 = fminf(fmaxf(x2, 0.f), W);
  float* o = boxes_out + (size_t)t * 4;
  o[0] = y1; o[1] = x1; o[2] = y2; o[3] = x2;
}

// -------- Kernel 2: per-(b,class) approx top-K + greedy NMS (1 block each) --------
__global__ __launch_bounds__(BDIM) void k_nms(PtrPack p, const float* boxes,
                                              float* sel_logit, float* sel_box) {
  __shared__ float s_logit[CAP];
  __shared__ int   s_idx[CAP];
  __shared__ float s_box[CAP * 4];
  __shared__ float s_stageA[CHUNK];
  __shared__ float s_stageB[CHUNK];
  __shared__ unsigned s_hist[NBINS];
  __shared__ int s_scan[BDIM];
  __shared__ unsigned long long s_red[BDIM / 32];
  __shared__ unsigned long long s_best;
  __shared__ int s_cnt;
  __shared__ int s_binT;

  const int tid = threadIdx.x;
  const int bc  = blockIdx.x;
  const int b   = bc / NCLS;
  const int c   = bc - b * NCLS;
  const int cls = c + 1;   // class 0 is background (dropped)

  for (int k = tid; k < NBINS; k += BDIM) s_hist[k] = 0u;
  if (tid == 0) s_cnt = 0;
  __syncthreads();

  const float BIN_SCALE = (float)NBINS / (10.0f - LOGIT_THR);

#define COLBASE(l_, j0_) (p.cls[(l_)] + (size_t)b * LVL_N[(l_)] * 91 + cls + (size_t)(j0_) * 91)

  // ---- pass 0: logit histogram over this class column (double-buffered TDM) ----
  {
    int l0, j00, r0;
    chunk_of(0, l0, j00, r0);
    stage_issue(COLBASE(l0, j00), r0, s_stageA, tid);
    for (int ci = 0; ci < NCHUNKS_TOT; ++ci) {
      float* cur = (ci & 1) ? s_stageB : s_stageA;
      float* nxt = (ci & 1) ? s_stageA : s_stageB;
      int l, j0, rows;
      chunk_of(ci, l, j0, rows);
      if (ci + 1 < NCHUNKS_TOT) {
        int l2, j02, r2;
        chunk_of(ci + 1, l2, j02, r2);
        stage_issue(COLBASE(l2, j02), r2, nxt, tid);
        stage_wait_keep1();     // in-order tensor ops: chunk ci is resident
      } else {
        stage_wait_all();
      }
      __syncthreads();
      for (int k = tid; k < rows; k += BDIM) {
        float L = cur[k];
        if (L >= LOGIT_THR) {
          int bin = (int)((L - LOGIT_THR) * BIN_SCALE);
          bin = bin < 0 ? 0 : (bin > NBINS - 1 ? NBINS - 1 : bin);
          atomicAdd(&s_hist[bin], 1u);
        }
      }
      __syncthreads();
    }
  }
  if (tid == 0) {
    unsigned acc = 0;
    int bt = 0;
    for (int q = NBINS - 1; q >= 0; --q) {
      acc += s_hist[q];
      if (acc > (unsigned)CAP) { bt = q; break; }
    }
    s_binT = bt;
  }
  __syncthreads();
  const int binT = s_binT;

  // ---- pass 1: deterministic compaction above refined threshold (pipelined) ----
  {
    int l0, j00, r0;
    chunk_of(0, l0, j00, r0);
    stage_issue(COLBASE(l0, j00), r0, s_stageA, tid);
    for (int ci = 0; ci < NCHUNKS_TOT; ++ci) {
      float* cur = (ci & 1) ? s_stageB : s_stageA;
      float* nxt = (ci & 1) ? s_stageA : s_stageB;
      int l, j0, rows;
      chunk_of(ci, l, j0, rows);
      if (ci + 1 < NCHUNKS_TOT) {
        int l2, j02, r2;
        chunk_of(ci + 1, l2, j02, r2);
        stage_issue(COLBASE(l2, j02), r2, nxt, tid);
        stage_wait_keep1();
      } else {
        stage_wait_all();
      }
      __syncthreads();
      int cl = 0;
      for (int k = tid; k < rows; k += BDIM) {
        float L = cur[k];
        if (L >= LOGIT_THR) {
          int bin = (int)((L - LOGIT_THR) * BIN_SCALE);
          bin = bin < 0 ? 0 : (bin > NBINS - 1 ? NBINS - 1 : bin);
          if (bin >= binT) cl++;
        }
      }
      s_scan[tid] = cl;
      __syncthreads();
      for (int off = 1; off < BDIM; off <<= 1) {  // inclusive scan
        int v = (tid >= off) ? s_scan[tid - off] : 0;
        __syncthreads();
        s_scan[tid] += v;
        __syncthreads();
      }
      int total = s_scan[BDIM - 1];
      int pos = s_cnt + s_scan[tid] - cl;         // exclusive base for this thread
      for (int k = tid; k < rows; k += BDIM) {
        float L = cur[k];
        if (L >= LOGIT_THR) {
          int bin = (int)((L - LOGIT_THR) * BIN_SCALE);
          bin = bin < 0 ? 0 : (bin > NBINS - 1 ? NBINS - 1 : bin);
          if (bin >= binT) {
            if (pos < CAP) { s_logit[pos] = L; s_idx[pos] = LVL_OFF[l] + j0 + k; }
            pos++;
          }
        }
      }
      __syncthreads();
      if (tid == 0) s_cnt += total;
      __syncthreads();
    }
  }
#undef COLBASE
  int cnt = s_cnt;
  if (cnt > CAP) cnt = CAP;

  // ---- gather candidate boxes into LDS (async global->LDS B128) ----
#if HAVE_ASYNC
  for (int k = tid; k < cnt; k += BDIM) {
    const float* src = boxes + ((size_t)b * NTOT + s_idx[k]) * 4;
    float* dst = &s_box[4 * k];
    __builtin_amdgcn_global_load_async_to_lds_b128(
        (as1_v4i*)(uintptr_t)src,
        (as3_v4i*)(unsigned)(uintptr_t)dst, 0, 0);
  }
  __builtin_amdgcn_s_wait_asynccnt(0);
  __syncthreads();
#else
  for (int k = tid; k < cnt; k += BDIM) {
    const float* src = boxes + ((size_t)b * NTOT + s_idx[k]) * 4;
    s_box[4 * k + 0] = src[0];
    s_box[4 * k + 1] = src[1];
    s_box[4 * k + 2] = src[2];
    s_box[4 * k + 3] = src[3];
  }
  __syncthreads();
#endif

  // ---- greedy NMS: argmax over active set each step (matches reference scan) ----
  float* my_sl = sel_logit + (size_t)bc * MAXT;
  float* my_sb = sel_box + (size_t)bc * MAXT * 4;
  for (int t = 0; t < MAXT; ++t) {
    unsigned long long key = 0ull;
    for (int k = tid; k < cnt; k += BDIM) {
      unsigned long long kk = ((unsigned long long)ordf(s_logit[k]) << 32) |
                              (unsigned)(0x7FFFFFFF - k);   // tie-break: lowest index
      if (kk > key) key = kk;
    }
    unsigned long long best = block_argmax(key, s_red, &s_best, tid);
    int ksel = 0x7FFFFFFF - (int)(best & 0xFFFFFFFFull);
    bool valid = (best != 0ull) && ksel >= 0 && ksel < cnt;
    float Ls = valid ? s_logit[ksel] : NEG_SENT;
    valid = valid && (Ls > -1.0e29f);
    float by1 = 0.f, bx1 = 0.f, by2 = 0.f, bx2 = 0.f;
    if (valid) {
      by1 = s_box[4 * ksel + 0];
      bx1 = s_box[4 * ksel + 1];
      by2 = s_box[4 * ksel + 2];
      bx2 = s_box[4 * ksel + 3];
    }
    __syncthreads();   // everyone done reading s_logit/s_box for this step
    if (valid) {
      float areaS = (by2 - by1) * (bx2 - bx1);
      for (int k = tid; k < cnt; k += BDIM) {
        if (k == ksel) { s_logit[k] = NEG_SENT; continue; }   // force-deactivate pick
        float L = s_logit[k];
        if (L <= -1.0e29f) continue;
        float cy1 = s_box[4 * k + 0], cx1 = s_box[4 * k + 1];
        float cy2 = s_box[4 * k + 2], cx2 = s_box[4 * k + 3];
        float yy1 = fmaxf(by1, cy1), xx1 = fmaxf(bx1, cx1);
        float yy2 = fminf(by2, cy2), xx2 = fminf(bx2, cx2);
        float inter = fmaxf(yy2 - yy1, 0.f) * fmaxf(xx2 - xx1, 0.f);
        float areaC = (cy2 - cy1) * (cx2 - cx1);
        float iou = inter / (areaS + areaC - inter + 1e-8f);
        if (iou > IOU_THR_F) s_logit[k] = NEG_SENT;
      }
    }
    if (tid == 0) {
      my_sl[t] = valid ? Ls : NEG_SENT;
      my_sb[4 * t + 0] = by1;
      my_sb[4 * t + 1] = bx1;
      my_sb[4 * t + 2] = by2;
      my_sb[4 * t + 3] = bx2;
    }
    __syncthreads();
  }
}

// -------- Kernel 3: per-batch final top-100 over 90*100 survivors --------
__global__ __launch_bounds__(BDIM) void k_final(const float* sel_logit,
                                                const float* sel_box, float* out) {
  __shared__ float s_sl[FLATN];
  __shared__ unsigned long long s_red[BDIM / 32];
  __shared__ unsigned long long s_best;
  const int tid = threadIdx.x;
  const int b = blockIdx.x;
  for (int k = tid; k < FLATN; k += BDIM)
    s_sl[k] = sel_logit[(size_t)b * FLATN + k];
  __syncthreads();

  float* ob = out;                          // [B,100,4]
  float* os = out + BATCH * MAXT * 4;       // [B,100]
  float* oc = os + BATCH * MAXT;            // [B,100] classes (as float)
  float* ov = oc + BATCH * MAXT;            // [B] valid counts
  int nvalid = 0;
  for (int t = 0; t < MAXT; ++t) {
    unsigned long long key = 0ull;
    for (int k = tid; k < FLATN; k += BDIM) {
      unsigned long long kk = ((unsigned long long)ordf(s_sl[k]) << 32) |
                              (unsigned)(0x7FFFFFFF - k);
      if (kk > key) key = kk;
    }
    unsigned long long best = block_argmax(key, s_red, &s_best, tid);
    int f = 0x7FFFFFFF - (int)(best & 0xFFFFFFFFull);
    if (f < 0 || f >= FLATN) f = 0;
    if (tid == 0) {
      float L = s_sl[f];
      bool valid = L > -1.0e29f;
      int o = b * MAXT + t;
      if (valid) {
        os[o] = 1.0f / (1.0f + __expf(-L));    // logit -> sigmoid score
        const float* bb = sel_box + ((size_t)b * FLATN + f) * 4;
        ob[4 * o + 0] = bb[0];
        ob[4 * o + 1] = bb[1];
        ob[4 * o + 2] = bb[2];
        ob[4 * o + 3] = bb[3];
        nvalid++;
      } else {
        os[o] = -1.0f;
        ob[4 * o + 0] = 0.f; ob[4 * o + 1] = 0.f;
        ob[4 * o + 2] = 0.f; ob[4 * o + 3] = 0.f;
      }
      oc[o] = (float)(f / MAXT + 1);
      s_sl[f] = NEG_SENT;
    }
    __syncthreads();
  }
  if (tid == 0) ov[b] = (float)nvalid;
}

extern "C" void kernel_launch(void* const* d_in, const int* in_sizes, int n_in,
                              void* d_out, int out_size, void* d_ws, size_t ws_size,
                              hipStream_t stream) {
  (void)in_sizes; (void)n_in; (void)out_size; (void)ws_size;
  PtrPack p;
  for (int l = 0; l < 5; ++l) {               // setup_inputs() dict order: interleaved per level
    p.box[l] = (const float*)d_in[l * 3 + 0];
    p.cls[l] = (const float*)d_in[l * 3 + 1];
    p.anc[l] = (const float*)d_in[l * 3 + 2];
  }
  const float* img = (const float*)d_in[15];

  float* ws = (float*)d_ws;
  float* ws_boxes = ws;                        // B*N*4 = 613,800 floats
  float* ws_sel_s = ws + 614400;               // 180*100 floats
  float* ws_sel_b = ws + 614400 + 18432;       // 180*100*4 floats
  float* out = (float*)d_out;

  k_decode<<<dim3((BATCH * NTOT + BDIM - 1) / BDIM), dim3(BDIM), 0, stream>>>(
      p, img, ws_boxes);
  k_nms<<<dim3(BATCH * NCLS), dim3(BDIM), 0, stream>>>(
      p, ws_boxes, ws_sel_s, ws_sel_b);
  k_final<<<dim3(BATCH), dim3(BDIM), 0, stream>>>(ws_sel_s, ws_sel_b, out);
}